// MemoryEfficientGATLayer_16123307229578
// MI455X (gfx1250) — compile-verified
//
#include <hip/hip_runtime.h>
#include <hip/hip_bf16.h>
#include <hip/hip_fp16.h>

typedef float v2f __attribute__((ext_vector_type(2)));
typedef float v8f __attribute__((ext_vector_type(8)));

#define IN_DIM 128
#define HDIM 128   // NUM_HEADS * OUT_DIM
#define NHEADS 4
#define ODIM 32

// LDS layout for B = fc_w^T, pair-interleaved + padded:
//   element (k, j) lives at float index (k>>1)*BPADF + j*2 + (k&1)
// -> the two K values a lane needs are one ds_load_b64;
// -> pair-row stride 272 floats (1088 B): the two lane-halves (pair p, p+1)
//    hit even-bank sets shifted by 16 -> disjoint banks, no conflicts.
#define BPADF 272
#define B_LDS_BYTES (64 * BPADF * 4)  // 64 k-pairs -> 69632 B (dynamic LDS)

// ---- ordered-uint encoding of float for atomicMax-based segment max ----
__device__ __forceinline__ unsigned fkey(float f) {
  unsigned b = __float_as_uint(f);
  return (b & 0x80000000u) ? ~b : (b | 0x80000000u);
}
__device__ __forceinline__ float funkey(unsigned k) {
  unsigned b = (k & 0x80000000u) ? (k & 0x7FFFFFFFu) : ~k;
  return __uint_as_float(b);
}
#define NEG_INF_KEY 0x007FFFFFu  // key(-inf)

// ---------------- Kernel 1: h = feat @ fc_w^T via V_WMMA_F32_16X16X4_F32 ---
// block = 256 threads (8 waves); block covers 128 rows; wave w owns rows
// [base + 16w, +16) and all 128 output columns (8 f32 accumulators).
__global__ __launch_bounds__(256) void gat_fc_wmma(
    const float* __restrict__ feat, const float* __restrict__ fc_w,
    float* __restrict__ h, int n) {
  extern __shared__ float Bs[];
  const int tid = threadIdx.x;
  // Stage fc_w^T into LDS (coalesced global reads; one-time scatter stores).
  for (int i = tid; i < IN_DIM * HDIM; i += 256) {
    const int kk = i & (IN_DIM - 1);  // consecutive lanes -> consecutive k
    const int jj = i >> 7;
    Bs[(kk >> 1) * BPADF + jj * 2 + (kk & 1)] = fc_w[jj * IN_DIM + kk];
  }
  __syncthreads();

  const int wave = tid >> 5;
  const int lane = tid & 31;
  const int half = lane >> 4;   // 0: K rows {k,k+1}; 1: K rows {k+2,k+3}
  const int l16  = lane & 15;
  const int rowbase = blockIdx.x * 128 + wave * 16;
  if (rowbase >= n) return;     // wave-uniform: EXEC stays all-ones for WMMA

  const int row = min(rowbase + l16, n - 1);
  const float* __restrict__ fr = feat + (size_t)row * IN_DIM + 2 * half;
  // Per-lane LDS base; all B loads are base + affine offset (immediates).
  const float* __restrict__ bbase = Bs + l16 * 2 + half * BPADF;

  v8f acc[8] = {};
  for (int kb = 0; kb < IN_DIM; kb += 16) {
    // A fragments for the 4 k-steps of this group (global_load_b64 each).
    v2f a[4];
#pragma unroll
    for (int s = 0; s < 4; ++s) a[s] = *(const v2f*)(fr + kb + 4 * s);

#pragma unroll
    for (int s = 0; s < 4; ++s) {
      const float* __restrict__ bk = bbase + ((kb >> 1) + 2 * s) * BPADF;
      // Batch all 8 B fragments (ds_load_b64, immediate offsets) ...
      v2f b[8];
#pragma unroll
      for (int t = 0; t < 8; ++t) b[t] = *(const v2f*)(bk + t * 32);
      // ... then issue the 8 WMMAs back-to-back.
#pragma unroll
      for (int t = 0; t < 8; ++t)
        acc[t] = __builtin_amdgcn_wmma_f32_16x16x4_f32(
            false, a[s], false, b[t], (short)0, acc[t], false, false);
    }
  }
#pragma unroll
  for (int j = 0; j < 8; ++j) {
    const int r = rowbase + j + 8 * half;  // D: M = j + 8*half, N = l16
    if (r < n) {
      float* hr = h + (size_t)r * HDIM;
#pragma unroll
      for (int t = 0; t < 8; ++t) hr[t * 16 + l16] = acc[t][j];
    }
  }
}

// ---------------- Kernel 0: init (out = feat residual; s = 0; m = -inf) ----
__global__ void gat_init(float* __restrict__ ssum, unsigned* __restrict__ mkey,
                         const float* __restrict__ feat, float* __restrict__ out,
                         int n) {
  int idx = blockIdx.x * blockDim.x + threadIdx.x;
  if (idx < n * NHEADS) {
    ssum[idx] = 0.0f;
    mkey[idx] = NEG_INF_KEY;
  }
  if (idx < n * (HDIM / 4)) {
    ((float4*)out)[idx] = ((const float4*)feat)[idx];
  }
}

// ---------------- Kernel 2: per-node attention logits ----------------------
__global__ void gat_node_logits(const float* __restrict__ h,
                                const float* __restrict__ attn_src,
                                const float* __restrict__ attn_dst,
                                float* __restrict__ e_src,
                                float* __restrict__ e_dst, int n) {
  int idx = blockIdx.x * blockDim.x + threadIdx.x;
  if (idx >= n * NHEADS) return;
  const int node = idx >> 2, hh = idx & 3;
  const float4* hp = (const float4*)(h + (size_t)node * HDIM + hh * ODIM);
  const float4* as = (const float4*)(attn_src + hh * ODIM);
  const float4* ad = (const float4*)(attn_dst + hh * ODIM);
  float ss = 0.0f, dd = 0.0f;
#pragma unroll
  for (int i = 0; i < ODIM / 4; ++i) {
    float4 v = hp[i], a = as[i], b = ad[i];
    ss += v.x * a.x + v.y * a.y + v.z * a.z + v.w * a.w;
    dd += v.x * b.x + v.y * b.y + v.z * b.z + v.w * b.w;
  }
  e_src[idx] = ss;
  e_dst[idx] = dd;
}

// ---------------- Kernel 3: per-edge logits + segment max ------------------
__global__ void gat_edge_logits(const int* __restrict__ src,
                                const int* __restrict__ dst,
                                const float* __restrict__ e_src,
                                const float* __restrict__ e_dst,
                                float* __restrict__ e_edge,
                                unsigned* __restrict__ mkey, int ne) {
  int idx = blockIdx.x * blockDim.x + threadIdx.x;
  if (idx >= ne * NHEADS) return;
  const int e = idx >> 2, hh = idx & 3;
  const int s = src[e], d = dst[e];
  float v = e_src[s * NHEADS + hh] + e_dst[d * NHEADS + hh];
  v = v > 0.0f ? v : 0.2f * v;  // LeakyReLU(0.2)
  e_edge[idx] = v;
  atomicMax(mkey + d * NHEADS + hh, fkey(v));
}

// ---------------- Kernel 4: exp(e - max) + segment sum ---------------------
__global__ void gat_edge_softmax(const int* __restrict__ dst,
                                 float* __restrict__ e_edge,
                                 const unsigned* __restrict__ mkey,
                                 float* __restrict__ ssum, int ne) {
  int idx = blockIdx.x * blockDim.x + threadIdx.x;
  if (idx >= ne * NHEADS) return;
  const int e = idx >> 2, hh = idx & 3;
  const int d = dst[e];
  const float m = funkey(mkey[d * NHEADS + hh]);
  const float a = __expf(e_edge[idx] - m);
  e_edge[idx] = a;  // reuse buffer as unnormalized attention
  unsafeAtomicAdd(ssum + d * NHEADS + hh, a);
}

// ---------------- Kernel 5: weighted scatter aggregation -------------------
// One wave per edge; lane covers 4 consecutive features (head = lane/8).
// out (51 MB) is L2-resident (192 MB), so the f32 atomics resolve in L2.
__global__ __launch_bounds__(256) void gat_aggregate(
    const int* __restrict__ src, const int* __restrict__ dst,
    const float* __restrict__ h, const float* __restrict__ a_edge,
    const float* __restrict__ ssum, float* __restrict__ out, int ne) {
  const int edge = blockIdx.x * 8 + (threadIdx.x >> 5);
  if (edge >= ne) return;
  const int lane = threadIdx.x & 31;
  const int s = src[edge], d = dst[edge];
  const int hh = lane >> 3;  // 4 floats of a lane lie within one 32-wide head
  const float a = a_edge[edge * NHEADS + hh];
  const float den = ssum[d * NHEADS + hh];
  const float w = a / (den > 0.0f ? den : 1.0f);
  const float4 hv = *(const float4*)(h + (size_t)s * HDIM + lane * 4);
  float* ob = out + (size_t)d * HDIM + lane * 4;
  unsafeAtomicAdd(ob + 0, hv.x * w);
  unsafeAtomicAdd(ob + 1, hv.y * w);
  unsafeAtomicAdd(ob + 2, hv.z * w);
  unsafeAtomicAdd(ob + 3, hv.w * w);
}

extern "C" void kernel_launch(void* const* d_in, const int* in_sizes, int n_in,
                              void* d_out, int out_size, void* d_ws, size_t ws_size,
                              hipStream_t stream) {
  const float* feat     = (const float*)d_in[0];
  const float* fc_w     = (const float*)d_in[1];
  const float* attn_src = (const float*)d_in[2];
  const float* attn_dst = (const float*)d_in[3];
  const int*   src      = (const int*)d_in[4];
  const int*   dst      = (const int*)d_in[5];
  float* out = (float*)d_out;

  const int n  = in_sizes[0] / IN_DIM;  // 100000
  const int ne = in_sizes[4];           // 1600000

  char* ws = (char*)d_ws;
  size_t off = 0;
  float*    h      = (float*)(ws + off); off += (size_t)n * HDIM * sizeof(float);
  float*    e_src  = (float*)(ws + off); off += (size_t)n * NHEADS * sizeof(float);
  float*    e_dst  = (float*)(ws + off); off += (size_t)n * NHEADS * sizeof(float);
  unsigned* mkey   = (unsigned*)(ws + off); off += (size_t)n * NHEADS * sizeof(unsigned);
  float*    ssum   = (float*)(ws + off); off += (size_t)n * NHEADS * sizeof(float);
  float*    e_edge = (float*)(ws + off); off += (size_t)ne * NHEADS * sizeof(float);
  (void)ws_size; (void)n_in; (void)out_size;

  const int T = 256;
  gat_init<<<(n * (HDIM / 4) + T - 1) / T, T, 0, stream>>>(ssum, mkey, feat, out, n);
  gat_fc_wmma<<<(n + 127) / 128, T, B_LDS_BYTES, stream>>>(feat, fc_w, h, n);
  gat_node_logits<<<(n * NHEADS + T - 1) / T, T, 0, stream>>>(h, attn_src, attn_dst,
                                                              e_src, e_dst, n);
  gat_edge_logits<<<(ne * NHEADS + T - 1) / T, T, 0, stream>>>(src, dst, e_src, e_dst,
                                                               e_edge, mkey, ne);
  gat_edge_softmax<<<(ne * NHEADS + T - 1) / T, T, 0, stream>>>(dst, e_edge, mkey,
                                                                ssum, ne);
  gat_aggregate<<<(ne + 7) / 8, T, 0, stream>>>(src, dst, h, e_edge, ssum, out, ne);
}